// BailingMoeV25_32633161515520
// MI455X (gfx1250) — compile-verified
//
#include <hip/hip_runtime.h>
#include <hip/hip_bf16.h>
#include <math.h>

// Problem constants (from reference setup_inputs)
constexpr int T  = 1024;
constexpr int H  = 2048;
constexpr int E  = 32;
constexpr int I  = 512;
constexpr int IS = 512;
constexpr int TOP_K = 4;
constexpr float ROUTED_SCALING = 2.5f;

typedef __attribute__((ext_vector_type(2))) float v2f;
typedef __attribute__((ext_vector_type(8))) float v8f;

__device__ __forceinline__ v8f wmma4(v2f a, v2f b, v8f c) {
  // D = A(16x4) * B(4x16) + C, full fp32
  return __builtin_amdgcn_wmma_f32_16x16x4_f32(false, a, false, b, (short)0, c,
                                               false, false);
}

// ---------------------------------------------------------------------------
// Kernel 1: router. One block per token, 256 threads.
// ---------------------------------------------------------------------------
__global__ __launch_bounds__(256) void router_kernel(
    const float* __restrict__ x,     // [T,H]
    const float* __restrict__ gw,    // [E,H]
    const float* __restrict__ bias,  // [E]
    float* __restrict__ Wrow,        // [T,E] dense combine weights
    int* __restrict__ counts,        // [E]
    int* __restrict__ toklist)       // [E,T]
{
  __shared__ float xs[H];
  __shared__ float part[256];
  const int t   = blockIdx.x;
  const int tid = threadIdx.x;

  // stage x row in LDS (coalesced float4)
  const float4* x4  = (const float4*)(x + (size_t)t * H);
  float4*       xs4 = (float4*)xs;
  for (int i = tid; i < H / 4; i += 256) xs4[i] = x4[i];
  __syncthreads();

  // 32 experts x 8 partial segments; threads j=0..7 of an expert read
  // consecutive float4s of gw (coalesced 128B groups).
  {
    const int e = tid >> 3, j = tid & 7;
    const float4* g4 = (const float4*)(gw + (size_t)e * H);
    float p = 0.f;
    for (int c = 0; c < (H / 4) / 8; ++c) {
      const int i4 = c * 8 + j;
      float4 xv = xs4[i4], gv = g4[i4];
      p += xv.x * gv.x + xv.y * gv.y + xv.z * gv.z + xv.w * gv.w;
    }
    part[tid] = p;
  }
  __syncthreads();

  if (tid < 32) {  // wave 0, lane == expert
    float logit = 0.f;
    #pragma unroll
    for (int j = 0; j < 8; ++j) logit += part[tid * 8 + j];
    const float score = 1.f / (1.f + __expf(-logit));   // sigmoid
    const float sc    = score + bias[tid];              // biased (selection)

    // group score = sum of top-2 biased scores within group of 8
    const int gbase = tid & ~7;
    float m1 = -INFINITY, m2 = -INFINITY;
    #pragma unroll
    for (int j = 0; j < 8; ++j) {
      float v = __shfl(sc, gbase + j, 32);
      if (v > m1)      { m2 = m1; m1 = v; }
      else if (v > m2) { m2 = v; }
    }
    const float gscore = m1 + m2;

    // top-2 of 4 groups (tie -> lower index, matches lax.top_k)
    float ga[4];
    ga[0] = __shfl(gscore, 0, 32);
    ga[1] = __shfl(gscore, 8, 32);
    ga[2] = __shfl(gscore, 16, 32);
    ga[3] = __shfl(gscore, 24, 32);
    int b1 = 0; float bv1 = ga[0];
    #pragma unroll
    for (int gi = 1; gi < 4; ++gi) if (ga[gi] > bv1) { bv1 = ga[gi]; b1 = gi; }
    int b2 = -1; float bv2 = -INFINITY;
    #pragma unroll
    for (int gi = 0; gi < 4; ++gi)
      if (gi != b1 && ga[gi] > bv2) { bv2 = ga[gi]; b2 = gi; }

    const int myg = tid >> 3;
    float mv = (myg == b1 || myg == b2) ? sc : -INFINITY;

    // top-4 experts across the wave (argmax butterfly, tie -> lower lane)
    int   sel[TOP_K];
    float selw[TOP_K];
    #pragma unroll
    for (int k = 0; k < TOP_K; ++k) {
      float v = mv; int ix = tid;
      #pragma unroll
      for (int off = 16; off > 0; off >>= 1) {
        float ov = __shfl_xor(v, off, 32);
        int   oi = __shfl_xor(ix, off, 32);
        if (ov > v || (ov == v && oi < ix)) { v = ov; ix = oi; }
      }
      sel[k]  = ix;
      selw[k] = __shfl(score, ix, 32);  // UNBIASED score as weight
      if (tid == ix) mv = -INFINITY;
    }
    const float wsum = selw[0] + selw[1] + selw[2] + selw[3];
    float w = 0.f;
    #pragma unroll
    for (int k = 0; k < TOP_K; ++k)
      if (tid == sel[k]) w = selw[k] / wsum;

    Wrow[(size_t)t * E + tid] = w;
    if (w > 0.f) {
      int pos = atomicAdd(&counts[tid], 1);
      toklist[tid * T + pos] = t;
    }
  }
}

// ---------------------------------------------------------------------------
// Kernel 1b: exclusive prefix-sum of counts -> offsets
// ---------------------------------------------------------------------------
__global__ void scan_kernel(const int* __restrict__ counts,
                            int* __restrict__ offsets) {
  if (threadIdx.x == 0) {
    int s = 0;
    for (int e = 0; e < E; ++e) { offsets[e] = s; s += counts[e]; }
    offsets[E] = s;
  }
}

// ---------------------------------------------------------------------------
// Kernel 2a: routed gate/up SwiGLU.
// Grid (T/32, I/64, E), block = 1 wave.  blockIdx.x = token tile so that
// launch-adjacent blocks share the same expert weight panel (L2 reuse).
// D-tile: 32 gathered tokens x 64 intermediate cols; each B fragment is
// reused by 2 WMMAs (two 16-row A fragments).
// ---------------------------------------------------------------------------
__global__ __launch_bounds__(32) void routed_gu_kernel(
    const float* __restrict__ x,        // [T,H]
    const float* __restrict__ wg,       // [E,H,I]
    const float* __restrict__ wu,       // [E,H,I]
    const float* __restrict__ Wrow,     // [T,E]
    const int* __restrict__ counts,
    const int* __restrict__ offsets,
    const int* __restrict__ toklist,    // [E,T]
    float* __restrict__ hbuf)           // [T*TOP_K, I] compacted
{
  const int e   = blockIdx.z;
  const int n_e = counts[e];
  const int mt  = blockIdx.x;
  if (mt * 32 >= n_e) return;           // uniform across block -> EXEC stays full
  const int i0   = blockIdx.y * 64;
  const int lane = threadIdx.x;
  const int half = lane >> 4;           // k-half
  const int r    = lane & 15;           // A row / B col

  const float* xrow[2];
  #pragma unroll
  for (int m = 0; m < 2; ++m) {
    const int slot = mt * 32 + m * 16 + r;
    const int cs   = (slot < n_e) ? slot : (n_e - 1);  // clamp (dup, not stored)
    xrow[m] = x + (size_t)toklist[e * T + cs] * H + 2 * half;
  }
  const float* bgp = wg + (size_t)e * H * I + i0 + r;
  const float* bup = wu + (size_t)e * H * I + i0 + r;

  v8f accg[2][4], accu[2][4];
  #pragma unroll
  for (int m = 0; m < 2; ++m)
    #pragma unroll
    for (int n = 0; n < 4; ++n)
      #pragma unroll
      for (int j = 0; j < 8; ++j) { accg[m][n][j] = 0.f; accu[m][n][j] = 0.f; }

  for (int k0 = 0; k0 < H; k0 += 4) {
    const v2f a0 = *(const v2f*)(xrow[0] + k0);
    const v2f a1 = *(const v2f*)(xrow[1] + k0);
    const float* bgk = bgp + (size_t)(k0 + 2 * half) * I;
    const float* buk = bup + (size_t)(k0 + 2 * half) * I;
    #pragma unroll
    for (int n = 0; n < 4; ++n) {
      v2f bg; bg.x = bgk[n * 16]; bg.y = bgk[I + n * 16];
      accg[0][n] = wmma4(a0, bg, accg[0][n]);
      accg[1][n] = wmma4(a1, bg, accg[1][n]);
      v2f bu; bu.x = buk[n * 16]; bu.y = buk[I + n * 16];
      accu[0][n] = wmma4(a0, bu, accu[0][n]);
      accu[1][n] = wmma4(a1, bu, accu[1][n]);
    }
  }

  // epilogue: h = silu(g) * u * combine_weight, compacted store
  const int obase = offsets[e];
  #pragma unroll
  for (int m = 0; m < 2; ++m) {
    #pragma unroll
    for (int v = 0; v < 8; ++v) {
      const int s2 = mt * 32 + m * 16 + v + 8 * half;
      if (s2 < n_e) {
        const int   tok = toklist[e * T + s2];
        const float wc  = Wrow[(size_t)tok * E + e];
        float* dst = hbuf + (size_t)(obase + s2) * I + i0 + r;
        #pragma unroll
        for (int n = 0; n < 4; ++n) {
          const float g = accg[m][n][v], u = accu[m][n][v];
          dst[n * 16] = (g / (1.f + __expf(-g))) * u * wc;
        }
      }
    }
  }
}

// ---------------------------------------------------------------------------
// Kernel 2b: routed down-proj, atomically accumulated (scaled by 2.5).
// Grid (T/32, H/64, E), block = 1 wave.
// ---------------------------------------------------------------------------
__global__ __launch_bounds__(32) void routed_down_kernel(
    const float* __restrict__ wd,       // [E,I,H]
    const int* __restrict__ counts,
    const int* __restrict__ offsets,
    const int* __restrict__ toklist,
    const float* __restrict__ hbuf,     // [T*TOP_K, I]
    float* __restrict__ out)            // [T,H] (holds shared-expert result)
{
  const int e   = blockIdx.z;
  const int n_e = counts[e];
  const int mt  = blockIdx.x;
  if (mt * 32 >= n_e) return;
  const int h0   = blockIdx.y * 64;
  const int lane = threadIdx.x;
  const int half = lane >> 4;
  const int r    = lane & 15;

  const float* arow[2];
  #pragma unroll
  for (int m = 0; m < 2; ++m) {
    const int slot = mt * 32 + m * 16 + r;
    const int cs   = (slot < n_e) ? slot : (n_e - 1);
    arow[m] = hbuf + (size_t)(offsets[e] + cs) * I + 2 * half;
  }
  const float* bp = wd + (size_t)e * I * H + h0 + r;

  v8f acc[2][4];
  #pragma unroll
  for (int m = 0; m < 2; ++m)
    #pragma unroll
    for (int n = 0; n < 4; ++n)
      #pragma unroll
      for (int j = 0; j < 8; ++j) acc[m][n][j] = 0.f;

  for (int k0 = 0; k0 < I; k0 += 4) {
    const v2f a0 = *(const v2f*)(arow[0] + k0);
    const v2f a1 = *(const v2f*)(arow[1] + k0);
    const float* bk = bp + (size_t)(k0 + 2 * half) * H;
    #pragma unroll
    for (int n = 0; n < 4; ++n) {
      v2f b; b.x = bk[n * 16]; b.y = bk[H + n * 16];
      acc[0][n] = wmma4(a0, b, acc[0][n]);
      acc[1][n] = wmma4(a1, b, acc[1][n]);
    }
  }

  #pragma unroll
  for (int m = 0; m < 2; ++m) {
    #pragma unroll
    for (int v = 0; v < 8; ++v) {
      const int s2 = mt * 32 + m * 16 + v + 8 * half;
      if (s2 < n_e) {
        const int tok = toklist[e * T + s2];
        float* dst = out + (size_t)tok * H + h0 + r;
        #pragma unroll
        for (int n = 0; n < 4; ++n)
          atomicAdd(dst + n * 16, ROUTED_SCALING * acc[m][n][v]);
      }
    }
  }
}

// ---------------------------------------------------------------------------
// Kernel 3a: shared-expert gate/up SwiGLU.  Grid (T/32, IS/64).
// ---------------------------------------------------------------------------
__global__ __launch_bounds__(32) void shared_gu_kernel(
    const float* __restrict__ x,     // [T,H]
    const float* __restrict__ wsg,   // [H,IS]
    const float* __restrict__ wsu,   // [H,IS]
    float* __restrict__ hs)          // [T,IS]
{
  const int mt   = blockIdx.x;
  const int i0   = blockIdx.y * 64;
  const int lane = threadIdx.x;
  const int half = lane >> 4;
  const int r    = lane & 15;

  const float* xrow[2];
  #pragma unroll
  for (int m = 0; m < 2; ++m)
    xrow[m] = x + (size_t)(mt * 32 + m * 16 + r) * H + 2 * half;
  const float* bgp = wsg + i0 + r;
  const float* bup = wsu + i0 + r;

  v8f accg[2][4], accu[2][4];
  #pragma unroll
  for (int m = 0; m < 2; ++m)
    #pragma unroll
    for (int n = 0; n < 4; ++n)
      #pragma unroll
      for (int j = 0; j < 8; ++j) { accg[m][n][j] = 0.f; accu[m][n][j] = 0.f; }

  for (int k0 = 0; k0 < H; k0 += 4) {
    const v2f a0 = *(const v2f*)(xrow[0] + k0);
    const v2f a1 = *(const v2f*)(xrow[1] + k0);
    const float* bgk = bgp + (size_t)(k0 + 2 * half) * IS;
    const float* buk = bup + (size_t)(k0 + 2 * half) * IS;
    #pragma unroll
    for (int n = 0; n < 4; ++n) {
      v2f bg; bg.x = bgk[n * 16]; bg.y = bgk[IS + n * 16];
      accg[0][n] = wmma4(a0, bg, accg[0][n]);
      accg[1][n] = wmma4(a1, bg, accg[1][n]);
      v2f bu; bu.x = buk[n * 16]; bu.y = buk[IS + n * 16];
      accu[0][n] = wmma4(a0, bu, accu[0][n]);
      accu[1][n] = wmma4(a1, bu, accu[1][n]);
    }
  }

  #pragma unroll
  for (int m = 0; m < 2; ++m) {
    #pragma unroll
    for (int v = 0; v < 8; ++v) {
      const int t = mt * 32 + m * 16 + v + 8 * half;
      float* dst = hs + (size_t)t * IS + i0 + r;
      #pragma unroll
      for (int n = 0; n < 4; ++n) {
        const float g = accg[m][n][v], u = accu[m][n][v];
        dst[n * 16] = (g / (1.f + __expf(-g))) * u;
      }
    }
  }
}

// ---------------------------------------------------------------------------
// Kernel 3b: shared-expert down-proj; plain store (initializes d_out).
// Grid (T/32, H/64).
// ---------------------------------------------------------------------------
__global__ __launch_bounds__(32) void shared_down_kernel(
    const float* __restrict__ hs,    // [T,IS]
    const float* __restrict__ wsd,   // [IS,H]
    float* __restrict__ out)         // [T,H]
{
  const int mt   = blockIdx.x;
  const int h0   = blockIdx.y * 64;
  const int lane = threadIdx.x;
  const int half = lane >> 4;
  const int r    = lane & 15;

  const float* arow[2];
  #pragma unroll
  for (int m = 0; m < 2; ++m)
    arow[m] = hs + (size_t)(mt * 32 + m * 16 + r) * IS + 2 * half;
  const float* bp = wsd + h0 + r;

  v8f acc[2][4];
  #pragma unroll
  for (int m = 0; m < 2; ++m)
    #pragma unroll
    for (int n = 0; n < 4; ++n)
      #pragma unroll
      for (int j = 0; j < 8; ++j) acc[m][n][j] = 0.f;

  for (int k0 = 0; k0 < IS; k0 += 4) {
    const v2f a0 = *(const v2f*)(arow[0] + k0);
    const v2f a1 = *(const v2f*)(arow[1] + k0);
    const float* bk = bp + (size_t)(k0 + 2 * half) * H;
    #pragma unroll
    for (int n = 0; n < 4; ++n) {
      v2f b; b.x = bk[n * 16]; b.y = bk[H + n * 16];
      acc[0][n] = wmma4(a0, b, acc[0][n]);
      acc[1][n] = wmma4(a1, b, acc[1][n]);
    }
  }

  #pragma unroll
  for (int m = 0; m < 2; ++m) {
    #pragma unroll
    for (int v = 0; v < 8; ++v) {
      const int t = mt * 32 + m * 16 + v + 8 * half;
      float* dst = out + (size_t)t * H + h0 + r;
      #pragma unroll
      for (int n = 0; n < 4; ++n) dst[n * 16] = acc[m][n][v];
    }
  }
}

// ---------------------------------------------------------------------------
// Workspace layout (bytes)
// ---------------------------------------------------------------------------
constexpr size_t OFF_W    = 0;                          // [T*E] float  = 128 KB
constexpr size_t OFF_CNT  = OFF_W + (size_t)T * E * 4;  // [E] int
constexpr size_t OFF_OFFS = OFF_CNT + 256;              // [E+1] int
constexpr size_t OFF_TOK  = OFF_OFFS + 256;             // [E*T] int    = 128 KB
constexpr size_t OFF_HB   = OFF_TOK + (size_t)E * T * 4;        // [T*4*I] f = 8 MB
constexpr size_t OFF_HS   = OFF_HB + (size_t)T * TOP_K * I * 4; // [T*IS]  f = 2 MB

extern "C" void kernel_launch(void* const* d_in, const int* in_sizes, int n_in,
                              void* d_out, int out_size, void* d_ws,
                              size_t ws_size, hipStream_t stream) {
  const float* x    = (const float*)d_in[0];
  const float* gw   = (const float*)d_in[1];
  const float* bias = (const float*)d_in[2];
  const float* wgt  = (const float*)d_in[3];
  const float* wup  = (const float*)d_in[4];
  const float* wdn  = (const float*)d_in[5];
  const float* wsg  = (const float*)d_in[6];
  const float* wsu  = (const float*)d_in[7];
  const float* wsd  = (const float*)d_in[8];
  float* out = (float*)d_out;

  char*  ws      = (char*)d_ws;
  float* Wrow    = (float*)(ws + OFF_W);
  int*   counts  = (int*)(ws + OFF_CNT);
  int*   offsets = (int*)(ws + OFF_OFFS);
  int*   toklist = (int*)(ws + OFF_TOK);
  float* hbuf    = (float*)(ws + OFF_HB);
  float* hs      = (float*)(ws + OFF_HS);

  hipMemsetAsync(counts, 0, E * sizeof(int), stream);

  router_kernel<<<T, 256, 0, stream>>>(x, gw, bias, Wrow, counts, toklist);
  scan_kernel<<<1, 32, 0, stream>>>(counts, offsets);

  // Shared expert first: writes d_out; routed path then atomically accumulates.
  shared_gu_kernel<<<dim3(T / 32, IS / 64), 32, 0, stream>>>(x, wsg, wsu, hs);
  shared_down_kernel<<<dim3(T / 32, H / 64), 32, 0, stream>>>(hs, wsd, out);

  routed_gu_kernel<<<dim3(T / 32, I / 64, E), 32, 0, stream>>>(
      x, wgt, wup, Wrow, counts, offsets, toklist, hbuf);
  routed_down_kernel<<<dim3(T / 32, H / 64, E), 32, 0, stream>>>(
      wdn, counts, offsets, toklist, hbuf, out);
}